// GNNLayer_3891240370406
// MI455X (gfx1250) — compile-verified
//
#include <hip/hip_runtime.h>
#include <hip/hip_bf16.h>
#include <math.h>

typedef float v2f __attribute__((ext_vector_type(2)));
typedef float v8f __attribute__((ext_vector_type(8)));

#define D_FEAT 128

// ---------------- init: zero c,z ; m = -inf bits ----------------
__global__ void init_state_kernel(float* __restrict__ c, float* __restrict__ z,
                                  unsigned int* __restrict__ mb, int V) {
    int i = blockIdx.x * blockDim.x + threadIdx.x;
    int n = V * D_FEAT;
    if (i < n) c[i] = 0.0f;
    if (i < V) { z[i] = 0.0f; mb[i] = 0xFF800000u; }  // -inf
}

// ---------------- transpose pack: W[R][C] -> WT[C][R] ----------------
__global__ void packT_kernel(const float* __restrict__ W, float* __restrict__ WT,
                             int R, int C) {
    int i = blockIdx.x * blockDim.x + threadIdx.x;
    if (i >= R * C) return;
    int n = i / C;
    int k = i - n * C;
    WT[(size_t)k * R + n] = W[i];
}

// ------- per-node dots with the two halves of W_edge (wave per node) -------
__global__ void node_dot_kernel(const float* __restrict__ nf,
                                const float* __restrict__ W_edge,
                                float* __restrict__ pd, float* __restrict__ ps,
                                int V) {
    int gid = blockIdx.x * blockDim.x + threadIdx.x;
    int v = gid >> 5;
    int lane = gid & 31;
    if (v >= V) return;
    float4 x  = ((const float4*)(nf + (size_t)v * D_FEAT))[lane];
    float4 w0 = ((const float4*)W_edge)[lane];
    float4 w1 = ((const float4*)(W_edge + D_FEAT))[lane];
    float d0 = x.x * w0.x + x.y * w0.y + x.z * w0.z + x.w * w0.w;
    float d1 = x.x * w1.x + x.y * w1.y + x.z * w1.z + x.w * w1.w;
#pragma unroll
    for (int off = 16; off > 0; off >>= 1) {
        d0 += __shfl_down(d0, off, 32);
        d1 += __shfl_down(d1, off, 32);
    }
    if (lane == 0) { pd[v] = d0; ps[v] = d1; }
}

// ---------------- f32 WMMA GEMM: Out[V,N] = X[V,128] @ B[128,N] + bias ------
// B is k-major (row-major [128][N]).  8 waves/block, 16 rows per block.
// A frag (16x4):  lane m = l%16, K = 2*(l/16)+r   -> float2
// B frag (4x16):  lane n = l%16, K = 2*(l/16)+r   -> float2
// C/D (16x16):    lane n = l%16, M = r + 8*(l/16) -> v8f
template <int N>
__global__ __launch_bounds__(256) void wmma_gemm_kernel(
    const float* __restrict__ X, const float* __restrict__ B,
    const float* __restrict__ bias, float* __restrict__ Out, int V) {
    __shared__ float xs[16][132];  // pad 132: 16 rows hit distinct banks
    const int tid = threadIdx.x;
    const int rowbase = blockIdx.x * 16;
#pragma unroll
    for (int i = 0; i < 8; ++i) {
        int idx = tid + i * 256;
        int r = idx >> 7;
        int col = idx & 127;
        int row = rowbase + r;
        if (row >= V) row = V - 1;
        xs[r][col] = X[(size_t)row * D_FEAT + col];
    }
    __syncthreads();
    const int lane = tid & 31;
    const int wave = tid >> 5;
    const int lo = lane & 15;
    const int hi = lane >> 4;
    const bool fullTile = (rowbase + 16 <= V);  // uniform across block
    for (int t = wave; t < N / 16; t += 8) {
        const int colbase = t * 16;
        v8f acc;
        const float bv = bias[colbase + lo];
#pragma unroll
        for (int r = 0; r < 8; ++r) acc[r] = bv;
#pragma unroll
        for (int kk = 0; kk < D_FEAT; kk += 4) {
            const int ka = kk + 2 * hi;
            v2f a, b;
            a[0] = xs[lo][ka];
            a[1] = xs[lo][ka + 1];
            b[0] = B[(size_t)ka * N + colbase + lo];
            b[1] = B[(size_t)(ka + 1) * N + colbase + lo];
            acc = __builtin_amdgcn_wmma_f32_16x16x4_f32(
                false, a, false, b, (short)0, acc, false, false);
        }
        if (fullTile) {
#pragma unroll
            for (int r = 0; r < 8; ++r) {
                int row = rowbase + r + 8 * hi;
                Out[(size_t)row * N + colbase + lo] = acc[r];
            }
        } else {
#pragma unroll
            for (int r = 0; r < 8; ++r) {
                int row = rowbase + r + 8 * hi;
                if (row < V) Out[(size_t)row * N + colbase + lo] = acc[r];
            }
        }
    }
}

// ------- edge pass 1: logit + atomic segment-max over dst ----------
__global__ void edge_logit_kernel(const int* __restrict__ src,
                                  const int* __restrict__ dst,
                                  const float* __restrict__ pd,
                                  const float* __restrict__ ps,
                                  const float* __restrict__ b_edge,
                                  float* __restrict__ logit,
                                  unsigned int* __restrict__ mb, int E) {
    int e = blockIdx.x * blockDim.x + threadIdx.x;
    if (e >= E) return;
    int d = dst[e];
    float x = pd[d] + ps[src[e]] + b_edge[0];
    float l = x > 0.0f ? x : 0.01f * x;  // leaky_relu
    logit[e] = l;
    int iv = __float_as_int(l);
    if (iv >= 0) atomicMax((int*)(mb + d), iv);
    else         atomicMin(mb + d, (unsigned int)iv);
}

// ------- edge pass 2 (wave/edge): e=exp(l-m); z[d]+=e; c[d,:]+=e*hv[s,:] ----
__global__ void edge_scatter_kernel(const int* __restrict__ src,
                                    const int* __restrict__ dst,
                                    const float* __restrict__ logit,
                                    const unsigned int* __restrict__ mb,
                                    float* __restrict__ z,
                                    const float* __restrict__ hv,
                                    float* __restrict__ c, int E) {
    int gid = blockIdx.x * blockDim.x + threadIdx.x;
    int e = gid >> 5;
    int lane = gid & 31;
    if (e >= E) return;
    int d = dst[e], s = src[e];
    float m = __int_as_float((int)mb[d]);
    float ev = expf(logit[e] - m);
    if (lane == 0) atomicAdd(z + d, ev);
    float4 h = ((const float4*)(hv + (size_t)s * D_FEAT))[lane];
    float* cd = c + (size_t)d * D_FEAT + lane * 4;
    atomicAdd(cd + 0, ev * h.x);
    atomicAdd(cd + 1, ev * h.y);
    atomicAdd(cd + 2, ev * h.z);
    atomicAdd(cd + 3, ev * h.w);
}

// ------- context = elu(c / z), in place ----------
__global__ void context_kernel(float* __restrict__ c, const float* __restrict__ z,
                               int V) {
    int i = blockIdx.x * blockDim.x + threadIdx.x;
    if (i >= V * D_FEAT) return;
    float zz = z[i >> 7];
    float x = zz > 0.0f ? c[i] / zz : 0.0f;
    c[i] = x > 0.0f ? x : expf(x) - 1.0f;
}

// ------- GRU gates + relu ----------
__global__ void gru_kernel(const float* __restrict__ gi, const float* __restrict__ gh,
                           const float* __restrict__ nf, float* __restrict__ out,
                           int V) {
    int i = blockIdx.x * blockDim.x + threadIdx.x;
    if (i >= V * D_FEAT) return;
    int v = i >> 7;
    int d = i & 127;
    size_t base = (size_t)v * 384;
    float ir = gi[base + d], iz = gi[base + 128 + d], in_ = gi[base + 256 + d];
    float hr = gh[base + d], hz = gh[base + 128 + d], hn  = gh[base + 256 + d];
    float r = 1.0f / (1.0f + expf(-(ir + hr)));
    float u = 1.0f / (1.0f + expf(-(iz + hz)));
    float n = tanhf(in_ + r * hn);
    float h = (1.0f - u) * n + u * nf[i];
    out[i] = h > 0.0f ? h : 0.0f;
}

extern "C" void kernel_launch(void* const* d_in, const int* in_sizes, int n_in,
                              void* d_out, int out_size, void* d_ws, size_t ws_size,
                              hipStream_t stream) {
    const float* nf     = (const float*)d_in[0];
    const int*   src    = (const int*)d_in[1];
    const int*   dst    = (const int*)d_in[2];
    const float* W_edge = (const float*)d_in[3];
    const float* b_edge = (const float*)d_in[4];
    const float* W_proj = (const float*)d_in[5];  // [128,128] k-major already
    const float* b_proj = (const float*)d_in[6];
    const float* W_ih   = (const float*)d_in[7];  // [384,128]
    const float* b_ih   = (const float*)d_in[8];
    const float* W_hh   = (const float*)d_in[9];  // [384,128]
    const float* b_hh   = (const float*)d_in[10];

    const int V = in_sizes[0] / D_FEAT;
    const int E = in_sizes[1];

    // workspace layout (floats)
    float* wsf = (float*)d_ws;
    float* pd    = wsf;
    float* ps    = pd + V;
    unsigned int* mb = (unsigned int*)(ps + V);
    float* z     = (float*)(mb + V);
    float* logit = z + V;
    float* hv    = logit + E;                    // [V,128]
    float* gh    = hv + (size_t)V * 128;         // [V,384]
    float* c     = gh + (size_t)V * 384;         // [V,128] -> context in place
    float* gi    = c + (size_t)V * 128;          // [V,384]
    float* WhhT  = gi + (size_t)V * 384;         // [128,384]
    float* WihT  = WhhT + 128 * 384;             // [128,384]

    const int blk = 256;

    // weight packing (transpose [384,128] -> [128,384])
    packT_kernel<<<(384 * 128 + blk - 1) / blk, blk, 0, stream>>>(W_hh, WhhT, 384, 128);
    packT_kernel<<<(384 * 128 + blk - 1) / blk, blk, 0, stream>>>(W_ih, WihT, 384, 128);

    // init accumulators
    init_state_kernel<<<((size_t)V * 128 + blk - 1) / blk, blk, 0, stream>>>(c, z, mb, V);

    // per-node edge-projection scalars
    node_dot_kernel<<<((size_t)V * 32 + blk - 1) / blk, blk, 0, stream>>>(
        nf, W_edge, pd, ps, V);

    // hv = nf @ W_proj + b_proj ; gh = nf @ W_hh^T + b_hh   (WMMA)
    const int gemmBlocks = (V + 15) / 16;
    wmma_gemm_kernel<128><<<gemmBlocks, 256, 0, stream>>>(nf, W_proj, b_proj, hv, V);
    wmma_gemm_kernel<384><<<gemmBlocks, 256, 0, stream>>>(nf, WhhT, b_hh, gh, V);

    // edge softmax passes
    edge_logit_kernel<<<(E + blk - 1) / blk, blk, 0, stream>>>(
        src, dst, pd, ps, b_edge, logit, mb, E);
    edge_scatter_kernel<<<((size_t)E * 32 + blk - 1) / blk, blk, 0, stream>>>(
        src, dst, logit, mb, z, hv, c, E);

    // context = elu(c / z)
    context_kernel<<<((size_t)V * 128 + blk - 1) / blk, blk, 0, stream>>>(c, z, V);

    // gi = context @ W_ih^T + b_ih   (WMMA)
    wmma_gemm_kernel<384><<<gemmBlocks, 256, 0, stream>>>(c, WihT, b_ih, gi, V);

    // GRU gates + relu -> output
    gru_kernel<<<((size_t)V * 128 + blk - 1) / blk, blk, 0, stream>>>(
        gi, gh, nf, (float*)d_out, V);
}